// GN_89266600280080
// MI455X (gfx1250) — compile-verified
//
#include <hip/hip_runtime.h>
#include <hip/hip_bf16.h>
#include <math.h>

// Problem constants (from reference): B=32768, O=32, D=1024
#define BATCH   32768
#define ODIM    32
#define DDIM    1024

// Tiling
#define KCHUNKS 64              // K split across grid
#define KPW     (BATCH/KCHUNKS) // 512 batch rows per WG
#define KHALF   (KPW/2)         // double-buffered halves of 256 rows (32 KB each)
#define NGROUPS 8               // D split across grid
#define NPW     (DDIM/NGROUPS)  // 128 columns per WG (16 per wave)
#define HFLOATS (KHALF*ODIM)    // 8192 floats per half

typedef __attribute__((ext_vector_type(2))) float v2f;
typedef __attribute__((ext_vector_type(8))) float v8f;

// Async-DMA one 32 KB half of the A operand (output chunk) into LDS, swizzled
// into the WMMA fragment layout at b32 granularity:
//   LDS float index of out[k][m]  =  ((ks*2 + hf)*32 + m)*2 + comp
//   with ks=k>>2, hf=(k>>1)&1, comp=k&1
// Per thread (m = tid&31, w = tid>>5) comp/hf are invariant, so both global and
// LDS addresses advance by a constant 1024 B per issue. Global side stays fully
// coalesced (consecutive lanes -> consecutive 4 B). Tracked on ASYNCcnt.
__device__ __forceinline__ void stage_async_half(const float* __restrict__ src,
                                                 const float* lds_dst, int tid)
{
    const int w    = tid >> 5;
    const int m    = tid & 31;
    const int comp = w & 1;
    const int hf   = (w >> 1) & 1;
    const int ksb  = w >> 2;

    const float* g = src + (size_t)tid;                        // 4 B per lane
    unsigned     l = (unsigned)(uintptr_t)lds_dst +
                     (unsigned)((((ksb * 2 + hf) * 32 + m) * 2 + comp) * 4);
    #pragma unroll
    for (int i = 0; i < HFLOATS / 256; ++i) {                  // 32 issues
        asm volatile("global_load_async_to_lds_b32 %0, %1, off"
                     :: "v"(l), "v"(g) : "memory");
        g += 256;                                              // +1024 B
        l += 1024;                                             // +1024 B
    }
}

// Phase 1: partial GEMM  part[c][o][d] = sum_{k in chunk c} out[k,o] * in[k,d]
// (weights + 1/B folded into phase 2).
// WMMA f32 16x16x4 operand layout (wave32):
//   A (16x4): lanes 0-15 M=lane {K0,K1}; lanes 16-31 M=lane-16 {K2,K3}
//   B (4x16): lanes 0-15 N=lane {K0,K1}; lanes 16-31 N=lane-16 {K2,K3}
__global__ void __launch_bounds__(256)
gn_gemm_partial(const float* __restrict__ outg,   // [B, 32]
                const float* __restrict__ inp,    // [B, 1024]
                float* __restrict__ part)         // [KCHUNKS, 32, 1024]
{
    __shared__ __attribute__((aligned(16))) float Abuf[2][HFLOATS]; // 64 KB

    const int tid    = threadIdx.x;
    const int bx     = blockIdx.x;
    const int chunk  = bx & (KCHUNKS - 1);
    const int ngrp   = bx >> 6;                   // bx / KCHUNKS
    const int wv     = tid >> 5;                  // wave id 0..7
    const int lane   = tid & 31;
    const int half   = lane >> 4;                 // 0: K{0,1}, 1: K{2,3}
    const int nl     = lane & 15;

    const int k0     = chunk * KPW;
    const int n0     = ngrp * NPW + wv * 16 + nl; // this lane's output column

    v8f c0 = {};   // M = 0..15
    v8f c1 = {};   // M = 16..31

    // Prologue: kick off DMA of half 0 (32 outstanding async ops max)
    stage_async_half(outg + (size_t)k0 * ODIM, &Abuf[0][0], tid);

    const float* bp_base = inp + (size_t)k0 * DDIM + n0;

    for (int ks2 = 0; ks2 < 2; ++ks2) {
        asm volatile("s_wait_asynccnt 0x0" ::: "memory");   // my half-ks2 ops done
        __syncthreads();                                    // everyone's done
        if (ks2 == 0) {
            // overlaps with the compute loop on Abuf[0] below
            stage_async_half(outg + (size_t)(k0 + KHALF) * ODIM, &Abuf[1][0], tid);
        }

        // ---- inner K loop: 64 steps of K=4, fed from LDS (A) + HBM stream (B) ----
        const float* bp = bp_base + (size_t)(ks2 * KHALF) * DDIM;
        #pragma unroll 4
        for (int kstep = 0; kstep < KHALF / 4; ++kstep) {
            // fragment-layout LDS: one ds_load_2addr_b64 covers both M-tiles
            const float* ap = &Abuf[ks2][((((kstep << 1) | half) << 5) + nl) * 2];
            v2f a0 = *(const v2f*)(ap);            // M-tile 0 (m = nl)
            v2f a1 = *(const v2f*)(ap + 32);       // M-tile 1 (m = 16+nl)
            v2f b;
            b.x = bp[(size_t)(2 * half)     * DDIM];
            b.y = bp[(size_t)(2 * half + 1) * DDIM];
            __builtin_prefetch(bp + (size_t)32 * DDIM, 0, 0);   // 32 rows (128 KB) ahead

            c0 = __builtin_amdgcn_wmma_f32_16x16x4_f32(false, a0, false, b,
                                                       (short)0, c0, false, false);
            c1 = __builtin_amdgcn_wmma_f32_16x16x4_f32(false, a1, false, b,
                                                       (short)0, c1, false, false);
            bp += (size_t)4 * DDIM;
        }
    }

    // ---- write partial C tile: C layout VGPR i -> M = i + 8*half ----
    float* pc = part + (size_t)chunk * (ODIM * DDIM) + n0;
    #pragma unroll 8
    for (int i = 0; i < 8; ++i) {
        pc[(size_t)(      8 * half + i) * DDIM] = c0[i];
        pc[(size_t)(16 +  8 * half + i) * DDIM] = c1[i];
    }
}

// Phase 2: gw[o] = sqrt( sum_d ( (w[o]/B) * sum_c part[c][o][d] )^2 )
__global__ void __launch_bounds__(256)
gn_rownorm(const float* __restrict__ part, const float* __restrict__ wts,
           float* __restrict__ gw)
{
    const int o   = blockIdx.x;
    const int tid = threadIdx.x;
    const float scale = wts[o] * (1.0f / (float)BATCH);

    float accum = 0.0f;
    for (int d = tid; d < DDIM; d += 256) {
        const float* p = part + (size_t)o * DDIM + d;
        float s = 0.0f;
        #pragma unroll 8
        for (int c = 0; c < KCHUNKS; ++c)
            s += p[(size_t)c * (ODIM * DDIM)];
        s *= scale;
        accum += s * s;
    }

    __shared__ float red[256];
    red[tid] = accum;
    __syncthreads();
    #pragma unroll
    for (int off = 128; off >= 1; off >>= 1) {
        if (tid < off) red[tid] += red[tid + off];
        __syncthreads();
    }
    if (tid == 0) gw[o] = sqrtf(red[0]);
}

extern "C" void kernel_launch(void* const* d_in, const int* in_sizes, int n_in,
                              void* d_out, int out_size, void* d_ws, size_t ws_size,
                              hipStream_t stream)
{
    const float* outg = (const float*)d_in[0];  // [B, 32] f32
    const float* inp  = (const float*)d_in[1];  // [B, 1024] f32
    const float* wts  = (const float*)d_in[2];  // [32] f32
    float*       gw   = (float*)d_out;          // [32] f32
    float*       part = (float*)d_ws;           // needs KCHUNKS*32*1024*4 = 8 MB

    gn_gemm_partial<<<dim3(KCHUNKS * NGROUPS), dim3(256), 0, stream>>>(outg, inp, part);
    gn_rownorm<<<dim3(ODIM), dim3(256), 0, stream>>>(part, wts, gw);
}